// MyLinear_9663676416464
// MI455X (gfx1250) — compile-verified
//
#include <hip/hip_runtime.h>

typedef float v2f __attribute__((ext_vector_type(2)));
typedef float v4f __attribute__((ext_vector_type(4)));
typedef float v8f __attribute__((ext_vector_type(8)));

#define IN_F 128
#define OUT_F 128
#define ROWS_PER_BLOCK 128      // 8 waves x 16 rows
#define THREADS 256
#define W_STRIDE 132            // 128 + 4 pad -> conflict-free b64 LDS reads

// Segment row counts (all multiples of 128, so blocks never straddle inputs)
#define N0 131072   // 32*64*64
#define N1 65536    // 32*32*64
#define N2 65536    // 32*64*32
#define N3 32768    // 32*32*32
#define NTOT (N0 + N1 + N2 + N3)   // 294912

__global__ __launch_bounds__(THREADS)
void grouped_linear_wmma_f32(const float* __restrict__ x0,
                             const float* __restrict__ x1,
                             const float* __restrict__ x2,
                             const float* __restrict__ x3,
                             const float* __restrict__ W,
                             const float* __restrict__ bias,
                             float* __restrict__ out)
{
    extern __shared__ float ldsW[];   // [128][W_STRIDE]

    const int tid  = threadIdx.x;
    const int lane = tid & 31;
    const int wave = tid >> 5;

    // ---- resolve which input segment this block's 128 rows live in (uniform) ----
    const long rowbase = (long)blockIdx.x * ROWS_PER_BLOCK;
    const float* src;
    long off;
    if (rowbase < N0)                { src = x0; off = rowbase; }
    else if (rowbase < N0 + N1)      { src = x1; off = rowbase - N0; }
    else if (rowbase < N0 + N1 + N2) { src = x2; off = rowbase - (N0 + N1); }
    else                             { src = x3; off = rowbase - (N0 + N1 + N2); }
    const float* xblk = src + (size_t)off * IN_F;

    // ---- stage W (row-major 128x128) into padded LDS with float4 stores ----
    // 4096 float4 elements total, 16 per thread.
    #pragma unroll
    for (int i = tid; i < (OUT_F * IN_F) / 4; i += THREADS) {
        const int row = i >> 5;        // /32 float4 per row
        const int c4  = i & 31;
        *(v4f*)(ldsW + row * W_STRIDE + c4 * 4) =
            *(const v4f*)(W + (size_t)row * IN_F + c4 * 4);
    }
    __syncthreads();

    // ---- per-lane operand geometry (ISA 32-bit 16x4 A / 4x16 B layout) ----
    const int m  = lane & 15;            // M index (A) / N index (B) / column (C,D)
    const int kk = (lane >> 4) << 1;     // lanes 0-15 carry K={0,1}, 16-31 carry K={2,3}

    // A: this wave's 16 rows, read straight from global (each row read exactly once)
    const float* aPtr = xblk + (size_t)(wave * 16 + m) * IN_F + kk;

    v8f acc[8];
    #pragma unroll
    for (int n = 0; n < 8; ++n) acc[n] = (v8f){0.f,0.f,0.f,0.f,0.f,0.f,0.f,0.f};

    // ---- K loop: 32 chunks of K=4; per chunk 1 global A load + 8 LDS B loads + 8 WMMAs ----
    for (int kc = 0; kc < IN_F / 4; ++kc) {
        const v2f a = *(const v2f*)(aPtr + kc * 4);
        const float* bBase = ldsW + kc * 4 + kk;   // + (n*16+m)*W_STRIDE selects W[n][k]
        #pragma unroll
        for (int n = 0; n < 8; ++n) {
            const v2f b = *(const v2f*)(bBase + (n * 16 + m) * W_STRIDE);
            acc[n] = __builtin_amdgcn_wmma_f32_16x16x4_f32(
                         false, a, false, b, (short)0, acc[n], false, false);
        }
    }

    // ---- epilogue: add bias (per output column), store to global ----
    // C/D layout: VGPR i -> row (i + 8*(lane>>4)), column = lane&15 within N tile.
    const size_t obase = (size_t)rowbase * OUT_F;
    const int rtop = wave * 16 + ((lane >> 4) << 3);
    #pragma unroll
    for (int n = 0; n < 8; ++n) {
        const float bv = bias[n * 16 + m];
        float* op = out + obase + (size_t)rtop * OUT_F + n * 16 + m;
        #pragma unroll
        for (int i = 0; i < 8; ++i) {
            op[(size_t)i * OUT_F] = acc[n][i] + bv;
        }
    }
}

extern "C" void kernel_launch(void* const* d_in, const int* in_sizes, int n_in,
                              void* d_out, int out_size, void* d_ws, size_t ws_size,
                              hipStream_t stream) {
    const float* x0   = (const float*)d_in[0];
    const float* x1   = (const float*)d_in[1];
    const float* x2   = (const float*)d_in[2];
    const float* x3   = (const float*)d_in[3];
    const float* W    = (const float*)d_in[4];
    const float* bias = (const float*)d_in[5];
    float* out = (float*)d_out;

    const int nblocks = NTOT / ROWS_PER_BLOCK;          // 2304
    const size_t smem = OUT_F * W_STRIDE * sizeof(float); // 67,584 B

    grouped_linear_wmma_f32<<<dim3(nblocks), dim3(THREADS), smem, stream>>>(
        x0, x1, x2, x3, W, bias, out);
}